// DiscreteEncoder_56865366999192
// MI455X (gfx1250) — compile-verified
//
#include <hip/hip_runtime.h>
#include <stdint.h>

#define NN 20000      // nodes
#define NE 320000     // edges
#define DIM 128
#define NCB 12000     // codebook entries
#define NG 256        // graphs
#define NL 5          // layers
#define CBP 132       // padded LDS row stride (floats): bank = (4*row + col) % 64

typedef float v2f __attribute__((ext_vector_type(2)));
typedef float v8f __attribute__((ext_vector_type(8)));
typedef __attribute__((address_space(3))) void lds_void_t;

__device__ __forceinline__ v8f wmma_f32(v2f a, v2f b, v8f c) {
  // V_WMMA_F32_16X16X4_F32 : D = A(16x4,f32) x B(4x16,f32) + C(16x16,f32)
  return __builtin_amdgcn_wmma_f32_16x16x4_f32(false, a, false, b, (short)0, c, false, false);
}

// per-lane async copy of 16 contiguous bytes global -> LDS (ASYNCcnt-tracked)
__device__ __forceinline__ void async_cp16(unsigned lds_byte_off, const float* g) {
  unsigned long long ga = (unsigned long long)(uintptr_t)g;
  asm volatile("global_load_async_to_lds_b128 %0, %1, off"
               :: "v"(lds_byte_off), "v"(ga) : "memory");
}
__device__ __forceinline__ void wait_async_le2() {
  asm volatile("s_wait_asynccnt 0x2" ::: "memory");
}
__device__ __forceinline__ void wait_async_0() {
  asm volatile("s_wait_asynccnt 0x0" ::: "memory");
}
__device__ __forceinline__ unsigned lds_offset_of(const void* p) {
  return (unsigned)(unsigned long long)(uintptr_t)(lds_void_t*)p;
}

// ---------------------------------------------------------------- utilities
__global__ void zero_f32(float* __restrict__ p, int n) {
  int i = blockIdx.x * blockDim.x + threadIdx.x;
  if (i < n) p[i] = 0.0f;
}

// ------------------------------------------------------------ edge scatter
// one wave per edge; each thread moves 4 contiguous features.
// agg is pre-seeded with h, so after this kernel agg = h + sum_{j->i} h_j.
__global__ void edge_agg(const float* __restrict__ h, const int* __restrict__ src,
                         const int* __restrict__ dst, float* __restrict__ agg) {
  long long tid = (long long)blockIdx.x * blockDim.x + threadIdx.x;
  int e = (int)(tid >> 5);
  if (e >= NE) return;
  int f4 = ((int)tid & 31) * 4;
  int s = src[e], d = dst[e];
  const float4 v = *(const float4*)(h + (long long)s * DIM + f4);
  float* o = agg + (long long)d * DIM + f4;
  atomicAdd(o + 0, v.x); atomicAdd(o + 1, v.y);
  atomicAdd(o + 2, v.z); atomicAdd(o + 3, v.w);
}

// ------------------------------------------------------------- WMMA GEMM
// C[M,128] = relu( A @ B[128,128] + bias ); 8 waves/block share W in LDS
__global__ __launch_bounds__(256) void gemm_mlp(
    const float* __restrict__ A, const float* __restrict__ B,
    const float* __restrict__ bias, float* __restrict__ C,
    int relu, int ntiles) {
  __shared__ float wt[DIM * CBP];          // W transposed: wt[n*CBP + k] = B[k*128+n]
  for (int idx = threadIdx.x; idx < DIM * DIM; idx += 256) {
    int k = idx >> 7, n = idx & 127;
    wt[n * CBP + k] = B[idx];
  }
  __syncthreads();

  int wave = threadIdx.x >> 5;
  int lane = threadIdx.x & 31;
  int m = lane & 15, half = lane >> 4;

  for (int tile = blockIdx.x * 8 + wave; tile < ntiles; tile += gridDim.x * 8) {
    long long row = (long long)tile * 16 + m;
    const float2* arow = (const float2*)(A + row * DIM);

    v8f acc[8];
    #pragma unroll
    for (int t = 0; t < 8; t++)
      #pragma unroll
      for (int q = 0; q < 8; q++) acc[t][q] = 0.0f;

    for (int kk = 0; kk < DIM; kk += 4) {
      float2 av = arow[(kk >> 1) + half];                 // A[m][kk+2h .. +1]
      v2f a; a[0] = av.x; a[1] = av.y;
      int kb = kk + 2 * half;
      #pragma unroll
      for (int t = 0; t < 8; t++) {
        int n = t * 16 + m;
        float2 bv = *(const float2*)&wt[n * CBP + kb];    // ds_load_b64, no conflicts
        v2f b; b[0] = bv.x; b[1] = bv.y;
        acc[t] = wmma_f32(a, b, acc[t]);
      }
    }
    #pragma unroll
    for (int t = 0; t < 8; t++) {
      #pragma unroll
      for (int v = 0; v < 8; v++) {
        int r = tile * 16 + v + 8 * half;                 // C layout: VGPR v -> row v / v+8
        int c = t * 16 + m;
        float val = acc[t][v] + bias[c];
        if (relu) val = fmaxf(val, 0.0f);
        C[(long long)r * DIM + c] = val;
      }
    }
  }
}

// -------------------------------------------------------------- BatchNorm
__global__ void bn_reduce(const float* __restrict__ z, float* __restrict__ sums,
                          float* __restrict__ sumsq) {
  int f = threadIdx.x;                 // 128 features
  int b = blockIdx.x;                  // 200 blocks x 100 rows
  const int rowsPer = NN / 200;
  float s = 0.0f, s2 = 0.0f;
  for (int r = b * rowsPer; r < (b + 1) * rowsPer; r++) {
    float v = z[(long long)r * DIM + f];
    s += v; s2 += v * v;
  }
  atomicAdd(&sums[f], s);
  atomicAdd(&sumsq[f], s2);
}

__global__ void bn_finalize(const float* __restrict__ sums, const float* __restrict__ sumsq,
                            const float* __restrict__ gamma, const float* __restrict__ beta,
                            float* __restrict__ scale, float* __restrict__ shift) {
  int f = threadIdx.x;
  float mu  = sums[f]  * (1.0f / NN);
  float var = sumsq[f] * (1.0f / NN) - mu * mu;
  float rs  = rsqrtf(var + 1e-5f);
  float a   = gamma[f] * rs;
  scale[f] = a;
  shift[f] = beta[f] - mu * a;
}

__global__ void bn_apply(const float* __restrict__ z, const float* __restrict__ scale,
                         const float* __restrict__ shift, float* __restrict__ h) {
  long long i = (long long)blockIdx.x * blockDim.x + threadIdx.x;
  if (i >= (long long)NN * DIM) return;
  int f = (int)(i & 127);
  h[i] = z[i] * scale[f] + shift[f];
}

// ------------------------------------------------------------ VQ: norms + argmin
__global__ void cb_norms(const float* __restrict__ cb, float* __restrict__ cc) {
  __shared__ float red[128];
  int j = blockIdx.x, f = threadIdx.x;
  float v = cb[(long long)j * DIM + f];
  red[f] = v * v;
  __syncthreads();
  for (int s = 64; s > 0; s >>= 1) { if (f < s) red[f] += red[f + s]; __syncthreads(); }
  if (f == 0) cc[j] = red[0];
}

// 8 waves/block, each owns a 16-node tile; codebook tiles double-buffered in LDS
// via async global->LDS copies. argmin_j (||c_j||^2 - 2 x.c_j), first-index ties.
__global__ __launch_bounds__(256) void vq_argmin(
    const float* __restrict__ x, const float* __restrict__ cb,
    const float* __restrict__ cc, int* __restrict__ bestidx) {
  __shared__ float cbt[2][16 * CBP];     // two 16x128 codebook tiles (padded rows)
  const unsigned lds_base = lds_offset_of(&cbt[0][0]);
  const unsigned buf_bytes = 16 * CBP * 4;

  int wave = threadIdx.x >> 5;
  int lane = threadIdx.x & 31;
  int m = lane & 15, half = lane >> 4;
  int tw   = blockIdx.x * 8 + wave;
  int tile = tw < (NN / 16) ? tw : (NN / 16 - 1);   // clamp; extra waves recompute

  // preload this wave's 16x128 A tile into registers
  const float2* xrow = (const float2*)(x + ((long long)tile * 16 + m) * DIM);
  float2 aA[32];
  #pragma unroll
  for (int w = 0; w < 32; w++) aA[w] = xrow[2 * w + half];   // A[m][4w+2h ..]

  float mind[8]; int mini[8];
  #pragma unroll
  for (int v = 0; v < 8; v++) { mind[v] = 3.4e38f; mini[v] = 0; }

  // stage tile 0 into buffer 0: 512 float4 chunks / 256 threads = 2 per thread
  {
    int t = threadIdx.x;
    #pragma unroll
    for (int i = 0; i < 2; i++) {
      int c = t + i * 256;            // 0..511
      int j = c >> 5, q = c & 31;     // row, float4-within-row
      async_cp16(lds_base + (unsigned)((j * CBP + q * 4) * 4),
                 cb + (long long)j * DIM + q * 4);
    }
  }

  const int NT = NCB / 16;            // 750 codebook tiles
  for (int jt = 0; jt < NT; jt++) {
    // prefetch next tile into the other buffer
    if (jt + 1 < NT) {
      int t = threadIdx.x;
      unsigned nb = lds_base + ((jt + 1) & 1) * buf_bytes;
      long long g0 = (long long)(jt + 1) * 16 * DIM;
      #pragma unroll
      for (int i = 0; i < 2; i++) {
        int c = t + i * 256;
        int j = c >> 5, q = c & 31;
        async_cp16(nb + (unsigned)((j * CBP + q * 4) * 4),
                   cb + g0 + (long long)j * DIM + q * 4);
      }
      wait_async_le2();               // current tile's 2 ops done (in-order completion)
    } else {
      wait_async_0();
    }
    __syncthreads();                  // all waves' staging for current tile landed

    const float* tb = &cbt[jt & 1][0];
    v8f acc;
    #pragma unroll
    for (int q = 0; q < 8; q++) acc[q] = 0.0f;
    #pragma unroll 8
    for (int w = 0; w < 32; w++) {
      int kb = 4 * w + 2 * half;
      float2 bv = *(const float2*)&tb[m * CBP + kb];        // ds_load_b64, no conflicts
      v2f a; a[0] = aA[w].x; a[1] = aA[w].y;
      v2f b; b[0] = bv.x; b[1] = bv.y;
      acc = wmma_f32(a, b, acc);
    }
    float ccj = cc[jt * 16 + m];      // this lane's column j
    #pragma unroll
    for (int v = 0; v < 8; v++) {
      float d = ccj - 2.0f * acc[v];
      if (d < mind[v]) { mind[v] = d; mini[v] = jt * 16 + m; }
    }
    __syncthreads();                  // consumption done before buffer is rewritten
  }

  // reduce argmin across the 16 lanes of each half (first-index tie-break)
  #pragma unroll
  for (int v = 0; v < 8; v++) {
    float mv = mind[v]; int mi = mini[v];
    for (int mask = 1; mask < 16; mask <<= 1) {
      float om = __shfl_xor(mv, mask, 32);
      int   oi = __shfl_xor(mi, mask, 32);
      if (om < mv || (om == mv && oi < mi)) { mv = om; mi = oi; }
    }
    if (m == 0 && tw < (NN / 16)) bestidx[tw * 16 + v + 8 * half] = mi;
  }
}

// ----------------------------------------------- quantize + commit + pooling
__global__ void pool_nodes(const float* __restrict__ x, const float* __restrict__ cb,
                           const int* __restrict__ bestidx, const float* __restrict__ score,
                           const int* __restrict__ batch, float* __restrict__ cgraph,
                           float* __restrict__ sgraph, float* __restrict__ cmt) {
  __shared__ float red[128];
  int i = blockIdx.x, f = threadIdx.x;
  float xv = x[(long long)i * DIM + f];
  float qv = cb[(long long)bestidx[i] * DIM + f];
  float diff = qv - xv;
  red[f] = diff * diff;
  float nr = xv + qv;                 // node_feat + q_st  (q_st == q numerically)
  float sc = score[i];
  int g = batch[i];
  atomicAdd(&cgraph[(long long)g * DIM + f], nr * sc);
  atomicAdd(&sgraph[(long long)g * DIM + f], nr * (1.0f - sc));
  __syncthreads();
  for (int s = 64; s > 0; s >>= 1) { if (f < s) red[f] += red[f + s]; __syncthreads(); }
  if (f == 0) atomicAdd(cmt, red[0]);
}

__global__ void write_cmt(const float* __restrict__ cmt, float* __restrict__ out) {
  out[0] = 0.25f * cmt[0] / (float)((long long)NN * DIM);
}

// ---------------------------------------------------------------- launcher
extern "C" void kernel_launch(void* const* d_in, const int* in_sizes, int n_in,
                              void* d_out, int out_size, void* d_ws, size_t ws_size,
                              hipStream_t stream) {
  const float* x        = (const float*)d_in[0];
  const int*   eidx     = (const int*)  d_in[1];
  const int*   batch    = (const int*)  d_in[2];
  const float* score    = (const float*)d_in[3];
  const float* conv_W1  = (const float*)d_in[4];
  const float* conv_b1  = (const float*)d_in[5];
  const float* conv_W2  = (const float*)d_in[6];
  const float* conv_b2  = (const float*)d_in[7];
  const float* bn_gamma = (const float*)d_in[8];
  const float* bn_beta  = (const float*)d_in[9];
  const float* codebook = (const float*)d_in[10];
  const float* fc1_W    = (const float*)d_in[11];
  const float* fc1_b    = (const float*)d_in[12];
  const float* fc2_W    = (const float*)d_in[13];
  const float* fc2_b    = (const float*)d_in[14];
  float* out = (float*)d_out;

  const long long ND = (long long)NN * DIM;          // 2,560,000
  float* ws     = (float*)d_ws;
  float* h      = ws;
  float* agg    = h    + ND;
  float* z1     = agg  + ND;
  float* z2     = z1   + ND;
  float* sums   = z2   + ND;        // 128
  float* sumsq  = sums  + 128;
  float* scale  = sumsq + 128;
  float* shift  = scale + 128;
  float* cc     = shift + 128;      // 12000
  float* cgraph = cc + NCB;         // 32768
  float* sgraph = cgraph + (long long)NG * DIM;
  float* cmt    = sgraph + (long long)NG * DIM;      // 1
  int*   bestix = (int*)(cmt + 1);                   // 20000

  const int* src = eidx;            // edge_index[0]
  const int* dst = eidx + NE;       // edge_index[1]

  const int NTILES = NN / 16;       // 1250
  const int GBLK   = (NTILES + 7) / 8;   // 157 blocks of 8 waves

  // h = x
  hipMemcpyAsync(h, x, ND * sizeof(float), hipMemcpyDeviceToDevice, stream);

  for (int l = 0; l < NL; l++) {
    // seed agg with h: after edge_agg, agg = h + sum_{j->i} h_j (GIN eps=0)
    hipMemcpyAsync(agg, h, ND * sizeof(float), hipMemcpyDeviceToDevice, stream);
    edge_agg<<<(NE * 32) / 256, 256, 0, stream>>>(h, src, dst, agg);
    // z1 = relu(agg @ W1 + b1)
    gemm_mlp<<<GBLK, 256, 0, stream>>>(agg, conv_W1 + (long long)l * DIM * DIM,
                                       conv_b1 + l * DIM, z1, 1, NTILES);
    // z2 = relu(z1 @ W2 + b2)
    gemm_mlp<<<GBLK, 256, 0, stream>>>(z1, conv_W2 + (long long)l * DIM * DIM,
                                       conv_b2 + l * DIM, z2, 1, NTILES);
    zero_f32<<<1, 256, 0, stream>>>(sums, 256);      // sums+sumsq contiguous
    bn_reduce<<<200, 128, 0, stream>>>(z2, sums, sumsq);
    bn_finalize<<<1, 128, 0, stream>>>(sums, sumsq, bn_gamma + l * DIM,
                                       bn_beta + l * DIM, scale, shift);
    bn_apply<<<(int)((ND + 255) / 256), 256, 0, stream>>>(z2, scale, shift, h);
  }

  // vector quantization
  cb_norms<<<NCB, 128, 0, stream>>>(codebook, cc);
  vq_argmin<<<GBLK, 256, 0, stream>>>(h, codebook, cc, bestix);

  // pooling accumulators (cgraph | sgraph | cmt contiguous)
  zero_f32<<<(NG * DIM * 2 + 1 + 255) / 256, 256, 0, stream>>>(cgraph, NG * DIM * 2 + 1);
  pool_nodes<<<NN, 128, 0, stream>>>(h, codebook, bestix, score, batch, cgraph, sgraph, cmt);

  // graph heads: out = [relu(cgraph@fc1+b1) | relu(sgraph@fc2+b2) | cmt]
  gemm_mlp<<<2, 256, 0, stream>>>(cgraph, fc1_W, fc1_b, out, 1, NG / 16);
  gemm_mlp<<<2, 256, 0, stream>>>(sgraph, fc2_W, fc2_b, out + NG * DIM, 1, NG / 16);
  write_cmt<<<1, 1, 0, stream>>>(cmt, out + 2 * NG * DIM);
}